// FeatDiversity_6760278524658
// MI455X (gfx1250) — compile-verified
//
#include <hip/hip_runtime.h>
#include <math.h>

// Problem constants from the reference: x:[B=9, N=4096, D=2048] f32, gamma=10.
constexpr int   Bq    = 9;
constexpr int   Nn    = 4096;
constexpr int   Dd    = 2048;
constexpr float GAMMA = 10.0f;

typedef float v2f __attribute__((ext_vector_type(2)));
typedef float v4f __attribute__((ext_vector_type(4)));
typedef float v8f __attribute__((ext_vector_type(8)));

// Workspace layout (floats): [0..255] snm accumulators, [256..2303] zero page.
constexpr int WS_FLOATS = 256 + Dd;

// ---------------------------------------------------------------------------
// Kernel 0: zero the accumulation slots and the zero page.
// ---------------------------------------------------------------------------
__global__ void fd_zero_kernel(float* __restrict__ ws) {
    const int i = blockIdx.x * 256 + threadIdx.x;
    if (i < WS_FLOATS) ws[i] = 0.0f;
}

// ---------------------------------------------------------------------------
// Kernel 1: per-position Gram via V_WMMA_F32_16X16X4_F32 + RBF + reduction.
// One wave per position n. For the symmetric Gram the A and B register images
// are identical, so one register pair serves both operands and the WMMA sums
// X[m][slot]*X[nn][slot] over the 4 (vgpr, half-wave) slots — the k-value each
// slot carries is arbitrary as long as it is covered exactly once. We exploit
// this: each lane loads a contiguous b128 (lane<16: k..k+3, lane>=16:
// k+4..k+7) whose .lo/.hi halves feed two WMMAs. Padded rows (9..15) read
// from a zeroed workspace page, so no masking VALU in the hot loop.
// ---------------------------------------------------------------------------
__global__ __launch_bounds__(256) void fd_gram_kernel(const float* __restrict__ x,
                                                      float* __restrict__ snm_acc,
                                                      const float* __restrict__ zeropage) {
    __shared__ float smem[256];
    const int tid = threadIdx.x;
    smem[tid] = 0.0f;
    __syncthreads();

    const int lane = tid & 31;
    const int wave = tid >> 5;
    const int n    = blockIdx.x * 8 + wave;          // 512 blocks * 8 waves = 4096

    const int row = lane & 15;                       // M (and N) index 0..15
    const int hi  = lane >> 4;                       // half-wave select
    const int koff = hi ? 4 : 0;                     // upper half-wave carries k+4..k+7

    const float* p = (row < Bq)
        ? (x + (size_t)row * ((size_t)Nn * Dd) + (size_t)n * Dd + koff)
        : (zeropage + koff);

    // Four independent accumulator chains; 8 WMMAs / 4 b128 loads per 32 k's.
    v8f c0 = {}, c1 = {}, c2 = {}, c3 = {};
    for (int k = 0; k < Dd; k += 32) {
        v4f b0 = *(const v4f*)(p + k);
        v4f b1 = *(const v4f*)(p + k + 8);
        v4f b2 = *(const v4f*)(p + k + 16);
        v4f b3 = *(const v4f*)(p + k + 24);
        c0 = __builtin_amdgcn_wmma_f32_16x16x4_f32(false, b0.lo, false, b0.lo, (short)0, c0, false, false);
        c1 = __builtin_amdgcn_wmma_f32_16x16x4_f32(false, b0.hi, false, b0.hi, (short)0, c1, false, false);
        c2 = __builtin_amdgcn_wmma_f32_16x16x4_f32(false, b1.lo, false, b1.lo, (short)0, c2, false, false);
        c3 = __builtin_amdgcn_wmma_f32_16x16x4_f32(false, b1.hi, false, b1.hi, (short)0, c3, false, false);
        c0 = __builtin_amdgcn_wmma_f32_16x16x4_f32(false, b2.lo, false, b2.lo, (short)0, c0, false, false);
        c1 = __builtin_amdgcn_wmma_f32_16x16x4_f32(false, b2.hi, false, b2.hi, (short)0, c1, false, false);
        c2 = __builtin_amdgcn_wmma_f32_16x16x4_f32(false, b3.lo, false, b3.lo, (short)0, c2, false, false);
        c3 = __builtin_amdgcn_wmma_f32_16x16x4_f32(false, b3.hi, false, b3.hi, (short)0, c3, false, false);
    }
    v8f c = (c0 + c1) + (c2 + c3);   // G[m][nn]: m = r + hi*8, nn = lane&15

    // Diagonal s[d] = G[d][d] = ||x_d||^2, broadcast to all lanes.
    // C layout: VGPR r, lanes 0-15 -> (M=r, N=lane); lanes 16-31 -> (M=r+8, N=lane-16).
    // diag d<8 lives at (VGPR d, lane d); diag d>=8 at (VGPR d-8, lane d+16).
    float sd[16];
#pragma unroll
    for (int d = 0; d < 8; ++d) {
        sd[d]     = __shfl(c[d], d, 32);
        sd[d + 8] = __shfl(c[d], d + 24, 32);
    }

    // s_nn for this lane's column index (constant-index cndmask chain).
    const int nn = row;
    float snn = sd[0];
#pragma unroll
    for (int d = 1; d < 16; ++d) snn = (nn == d) ? sd[d] : snn;

    // RBF + mask + block-level LDS reduction.
#pragma unroll
    for (int r = 0; r < 8; ++r) {
        const int   m  = r + (hi << 3);
        const float sm = hi ? sd[r + 8] : sd[r];
        float sq = sm + snn - 2.0f * c[r];
        sq = fmaxf(sq, 0.0f);
        const bool ok = (m < Bq) && (nn < Bq) && (m != nn);
        if (ok) {
            const float e = __expf(-GAMMA * sq);
            atomicAdd(&smem[m * 16 + nn], e);      // ds_add_f32
        }
    }
    __syncthreads();

    atomicAdd(&snm_acc[tid], smem[tid]);           // global_atomic_add_f32, 256/block
}

// ---------------------------------------------------------------------------
// Kernel 2: single-thread finalize. Build snm (9x9, mean over N, zero diag),
// then direct sum + double-precision LU with partial pivoting for det/logdet.
// ---------------------------------------------------------------------------
__global__ void fd_finalize_kernel(const float* __restrict__ snm_acc,
                                   float* __restrict__ out) {
    if (threadIdx.x != 0 || blockIdx.x != 0) return;

    double M[Bq][Bq];
    double direct = 0.0;
    const double invN = 1.0 / (double)Nn;
    for (int i = 0; i < Bq; ++i) {
        for (int j = 0; j < Bq; ++j) {
            double v = (i == j) ? 0.0 : (double)snm_acc[i * 16 + j] * invN;
            M[i][j] = v;
            direct += v;
        }
    }

    // LU with partial pivoting; det = (+-1) * product of pivots.
    double det = 1.0;
    for (int k = 0; k < Bq; ++k) {
        int piv = k;
        double best = fabs(M[k][k]);
        for (int i = k + 1; i < Bq; ++i) {
            double a = fabs(M[i][k]);
            if (a > best) { best = a; piv = i; }
        }
        if (piv != k) {
            for (int j = 0; j < Bq; ++j) {
                double t = M[k][j]; M[k][j] = M[piv][j]; M[piv][j] = t;
            }
            det = -det;
        }
        const double pk = M[k][k];
        det *= pk;
        if (pk != 0.0) {
            for (int i = k + 1; i < Bq; ++i) {
                double f = M[i][k] / pk;
                for (int j = k; j < Bq; ++j) M[i][j] -= f * M[k][j];
            }
        }
    }

    out[0] = (float)direct;                              // direct_div
    out[1] = (float)(-det);                              // det_div
    out[2] = (det > 0.0) ? (float)(-log(det))            // logdet_div
                         : __int_as_float(0x7FC00000);   // NaN for non-positive det
}

// ---------------------------------------------------------------------------
extern "C" void kernel_launch(void* const* d_in, const int* in_sizes, int n_in,
                              void* d_out, int out_size, void* d_ws, size_t ws_size,
                              hipStream_t stream) {
    (void)in_sizes; (void)n_in; (void)out_size; (void)ws_size;
    const float* x   = (const float*)d_in[0];
    float*       out = (float*)d_out;
    float*       ws  = (float*)d_ws;

    fd_zero_kernel<<<(WS_FLOATS + 255) / 256, 256, 0, stream>>>(ws);
    fd_gram_kernel<<<Nn / 8, 256, 0, stream>>>(x, ws, ws + 256);
    fd_finalize_kernel<<<1, 1, 0, stream>>>(ws, out);
}